// AttnBottleneck_77704548319989
// MI455X (gfx1250) — compile-verified
//
#include <hip/hip_runtime.h>
#include <hip/hip_bf16.h>
#include <cstddef>

// Problem constants (from reference)
#define HD   5120      // HIDDEN_DIM
#define BOT  64        // BOTTLENECK
#define NW   17        // N_WEEKS
#define BB   128       // batch
#define SS   512       // sequence
#define NC   8         // S-chunks per batch row in fused path (64 rows each)

typedef __attribute__((ext_vector_type(2))) float v2f;
typedef __attribute__((ext_vector_type(8))) float v8f;

// ---------------------------------------------------------------------------
// Block-wide (256 threads = 8 wave32) sum reduction, result broadcast to all.
// ---------------------------------------------------------------------------
__device__ __forceinline__ float block_reduce_sum_256(float v, float* red, float* bcast) {
    const int t = threadIdx.x, wid = t >> 5, lane = t & 31;
#pragma unroll
    for (int off = 16; off > 0; off >>= 1) v += __shfl_xor(v, off, 32);
    if (lane == 0) red[wid] = v;
    __syncthreads();
    if (wid == 0) {
        float r = (lane < 8) ? red[lane] : 0.f;
#pragma unroll
        for (int off = 4; off > 0; off >>= 1) r += __shfl_xor(r, off, 32);
        if (lane == 0) *bcast = r;
    }
    __syncthreads();
    return *bcast;
}

// ===========================================================================
// FUSED PATH (single pass over x): per (b, s-chunk) block computes raw scores
// AND an online-softmax partial pooled vector in one read of its x slice.
// ===========================================================================
__global__ __launch_bounds__(256)
void fused_attn_kernel(const float* __restrict__ x, const float* __restrict__ aw,
                       float* __restrict__ scores, float* __restrict__ part,
                       float* __restrict__ cmax, float* __restrict__ csum) {
    __shared__ float red[8];
    __shared__ float bcast;
    const int b = blockIdx.y;        // batch row
    const int c = blockIdx.x;        // s-chunk (64 rows)
    const int t = threadIdx.x;       // 0..255; owns d4 = t + k*256, k=0..4

    // attn_w slice cached in registers
    const float4* __restrict__ aw4 = (const float4*)aw;
    float4 w4[5];
#pragma unroll
    for (int k = 0; k < 5; ++k) w4[k] = aw4[t + k * 256];

    const float4* __restrict__ xr =
        (const float4*)(x + ((size_t)b * SS + (size_t)c * (SS / NC)) * HD);

    float4 acc[5];
#pragma unroll
    for (int k = 0; k < 5; ++k) acc[k] = make_float4(0.f, 0.f, 0.f, 0.f);
    float m = -INFINITY, sumexp = 0.f;

    // software pipeline: preload row 0
    float4 cur[5];
#pragma unroll
    for (int k = 0; k < 5; ++k) cur[k] = xr[t + k * 256];

    for (int s = 0; s < SS / NC; ++s) {
        // partial dot for the score
        float p = 0.f;
#pragma unroll
        for (int k = 0; k < 5; ++k)
            p += cur[k].x * w4[k].x + cur[k].y * w4[k].y
               + cur[k].z * w4[k].z + cur[k].w * w4[k].w;

        // prefetch next row before the barrier-bearing reduction
        const int sn = (s + 1 < SS / NC) ? s + 1 : s;
        float4 nxt[5];
#pragma unroll
        for (int k = 0; k < 5; ++k)
            nxt[k] = xr[(size_t)sn * (HD / 4) + t + k * 256];

        const float score = block_reduce_sum_256(p, red, &bcast);
        if (t == 0) scores[b * SS + c * (SS / NC) + s] = score;

        // online softmax update (block-uniform branch; rescale ~log(64) times)
        float e;
        if (score > m) {
            const float scale = expf(m - score);   // exp(-inf)=0 on first row
#pragma unroll
            for (int k = 0; k < 5; ++k) {
                acc[k].x *= scale; acc[k].y *= scale;
                acc[k].z *= scale; acc[k].w *= scale;
            }
            sumexp *= scale;
            m = score;
            e = 1.f;
        } else {
            e = expf(score - m);
        }
        sumexp += e;
#pragma unroll
        for (int k = 0; k < 5; ++k) {
            acc[k].x += e * cur[k].x; acc[k].y += e * cur[k].y;
            acc[k].z += e * cur[k].z; acc[k].w += e * cur[k].w;
        }
#pragma unroll
        for (int k = 0; k < 5; ++k) cur[k] = nxt[k];
    }

    float4* pp = (float4*)(part + ((size_t)b * NC + c) * HD);
#pragma unroll
    for (int k = 0; k < 5; ++k) pp[t + k * 256] = acc[k];
    if (t == 0) { cmax[b * NC + c] = m; csum[b * NC + c] = sumexp; }
}

// Combine the NC chunk partials per batch row; emit final w and pooled.
__global__ __launch_bounds__(256)
void finalize_kernel(const float* __restrict__ scores, const float* __restrict__ part,
                     const float* __restrict__ cmax, const float* __restrict__ csum,
                     float* __restrict__ w_out, float* __restrict__ pooled) {
    const int b = blockIdx.x;
    const int t = threadIdx.x;

    // redundant per-thread scan of 8 (max, sumexp) pairs (L2-resident, trivial)
    float M = -INFINITY;
#pragma unroll
    for (int c = 0; c < NC; ++c) M = fmaxf(M, cmax[b * NC + c]);
    float T = 0.f;
    float cw[NC];
#pragma unroll
    for (int c = 0; c < NC; ++c) {
        const float f = expf(cmax[b * NC + c] - M);
        T += csum[b * NC + c] * f;
        cw[c] = f;
    }
    const float invT = 1.f / T;
#pragma unroll
    for (int c = 0; c < NC; ++c) cw[c] *= invT;

    // softmax weights output (returned tensor)
#pragma unroll
    for (int i = 0; i < SS / 256; ++i) {
        const int s = t + i * 256;
        w_out[b * SS + s] = expf(scores[b * SS + s] - M) * invT;
    }

    // pooled[b,d] = sum_c cw[c] * part[b,c,d]
    const float4* pb = (const float4*)(part + (size_t)b * NC * HD);
    float4* po = (float4*)(pooled + (size_t)b * HD);
#pragma unroll
    for (int k = 0; k < 5; ++k) {
        const int d4 = t + k * 256;
        float4 a = make_float4(0.f, 0.f, 0.f, 0.f);
#pragma unroll
        for (int c = 0; c < NC; ++c) {
            float4 v = pb[(size_t)c * (HD / 4) + d4];
            a.x += cw[c] * v.x; a.y += cw[c] * v.y;
            a.z += cw[c] * v.z; a.w += cw[c] * v.w;
        }
        po[d4] = a;
    }
}

// ===========================================================================
// FALLBACK PATH (two passes over x) — used when workspace is too small.
// ===========================================================================
__global__ __launch_bounds__(256)
void scores_kernel(const float* __restrict__ x, const float* __restrict__ aw,
                   float* __restrict__ scores) {
    const int row  = blockIdx.x * 8 + (threadIdx.x >> 5);
    const int lane = threadIdx.x & 31;
    const float4* __restrict__ xr  = (const float4*)(x + (size_t)row * HD);
    const float4* __restrict__ aw4 = (const float4*)aw;
    float acc = 0.f;
#pragma unroll 8
    for (int i = lane; i < HD / 4; i += 32) {
        float4 a = xr[i];
        float4 b = aw4[i];
        acc += a.x * b.x + a.y * b.y + a.z * b.z + a.w * b.w;
    }
#pragma unroll
    for (int off = 16; off > 0; off >>= 1) acc += __shfl_xor(acc, off, 32);
    if (lane == 0) scores[row] = acc;
}

__global__ __launch_bounds__(512)
void softmax_kernel(const float* __restrict__ scores,
                    float* __restrict__ w_ws, float* __restrict__ w_out) {
    __shared__ float red[16];
    __shared__ float gmax_s, gsum_s;
    const int b = blockIdx.x, t = threadIdx.x;
    const int wid = t >> 5, lane = t & 31;

    float s = scores[b * SS + t];
    float m = s;
#pragma unroll
    for (int off = 16; off > 0; off >>= 1) m = fmaxf(m, __shfl_xor(m, off, 32));
    if (lane == 0) red[wid] = m;
    __syncthreads();
    if (t < 16) {
        float v = red[t];
#pragma unroll
        for (int off = 8; off > 0; off >>= 1) v = fmaxf(v, __shfl_xor(v, off, 32));
        if (t == 0) gmax_s = v;
    }
    __syncthreads();

    const float e = expf(s - gmax_s);
    float sum = e;
#pragma unroll
    for (int off = 16; off > 0; off >>= 1) sum += __shfl_xor(sum, off, 32);
    __syncthreads();
    if (lane == 0) red[wid] = sum;
    __syncthreads();
    if (t < 16) {
        float v = red[t];
#pragma unroll
        for (int off = 8; off > 0; off >>= 1) v += __shfl_xor(v, off, 32);
        if (t == 0) gsum_s = v;
    }
    __syncthreads();

    const float wv = e / gsum_s;
    w_ws [b * SS + t] = wv;
    w_out[b * SS + t] = wv;
}

__global__ __launch_bounds__(256)
void pooled_kernel(const float* __restrict__ x, const float* __restrict__ w_ws,
                   float* __restrict__ pooled) {
    __shared__ float wl[SS];
    const int b = blockIdx.y;
    const int t = threadIdx.x;
    wl[t]       = w_ws[b * SS + t];
    wl[t + 256] = w_ws[b * SS + t + 256];
    __syncthreads();

    const int d4 = blockIdx.x * 256 + t;
    const float4* __restrict__ xr = (const float4*)(x + (size_t)b * SS * HD);
    float4 acc = make_float4(0.f, 0.f, 0.f, 0.f);
#pragma unroll 4
    for (int s = 0; s < SS; ++s) {
        float4 v = xr[(size_t)s * (HD / 4) + d4];
        float  ww = wl[s];
        acc.x += ww * v.x; acc.y += ww * v.y;
        acc.z += ww * v.z; acc.w += ww * v.w;
    }
    ((float4*)(pooled + (size_t)b * HD))[d4] = acc;
}

// ===========================================================================
// Shared tail: WMMA fp32 GEMM (h = relu(pooled @ w1 + b1)) and tiny out GEMM.
// ===========================================================================
__global__ __launch_bounds__(32)
void mlp1_wmma_kernel(const float* __restrict__ pooled, const float* __restrict__ w1,
                      const float* __restrict__ b1, float* __restrict__ h) {
    const int mt   = blockIdx.x;          // 0..7
    const int nt   = blockIdx.y;          // 0..3
    const int lane = threadIdx.x;         // 0..31
    const int half = lane >> 4;
    const int i16  = lane & 15;

    const int m = mt * 16 + i16;
    const int n = nt * 16 + i16;

    const float* __restrict__ arow = pooled + (size_t)m * HD + 2 * half;
    const float* __restrict__ bcol = w1 + (size_t)(2 * half) * BOT + n;

    v8f c0 = {}, c1 = {}, c2 = {}, c3 = {};
    for (int k = 0; k < HD; k += 16) {
#pragma unroll
        for (int u = 0; u < 4; ++u) {
            const int kk = k + 4 * u;
            v2f a = *(const v2f*)(arow + kk);
            v2f bb;
            bb.x = bcol[(size_t)kk * BOT];
            bb.y = bcol[(size_t)kk * BOT + BOT];
            v8f& c = (u == 0) ? c0 : (u == 1) ? c1 : (u == 2) ? c2 : c3;
            c = __builtin_amdgcn_wmma_f32_16x16x4_f32(
                    false, a, false, bb, (short)0, c, false, false);
        }
    }
    v8f cs = c0 + c1 + c2 + c3;

    const float bias = b1[n];
#pragma unroll
    for (int r = 0; r < 8; ++r) {
        const int row = mt * 16 + r + 8 * half;
        h[(size_t)row * BOT + n] = fmaxf(cs[r] + bias, 0.f);
    }
}

__global__ __launch_bounds__(32)
void mlp2_kernel(const float* __restrict__ h, const float* __restrict__ w2,
                 const float* __restrict__ b2, float* __restrict__ out) {
    const int mb = blockIdx.x;
    const int j  = threadIdx.x;
    if (j < NW) {
        float acc = b2[j];
        const float* hr = h + (size_t)mb * BOT;
#pragma unroll 8
        for (int n = 0; n < BOT; ++n) acc += hr[n] * w2[n * NW + j];
        out[mb * NW + j] = acc;
    }
}

// ---------------------------------------------------------------------------
extern "C" void kernel_launch(void* const* d_in, const int* in_sizes, int n_in,
                              void* d_out, int out_size, void* d_ws, size_t ws_size,
                              hipStream_t stream) {
    const float* x  = (const float*)d_in[0];   // [128,512,5120]
    const float* aw = (const float*)d_in[1];   // [5120]
    const float* w1 = (const float*)d_in[2];   // [5120,64]
    const float* b1 = (const float*)d_in[3];   // [64]
    const float* w2 = (const float*)d_in[4];   // [64,17]
    const float* b2 = (const float*)d_in[5];   // [17]
    (void)in_sizes; (void)n_in; (void)out_size;

    float* out_p = (float*)d_out;              // [128*17]
    float* w_out = out_p + BB * NW;            // [128*512]
    float* ws    = (float*)d_ws;

    // fused-path workspace: scores | cmax | csum | part | pooled | h  (~24 MB)
    const size_t F_SC = (size_t)BB * SS;           // 65536
    const size_t F_CM = 1024, F_CS = 1024;
    const size_t F_PT = (size_t)BB * NC * HD;      // 5,242,880
    const size_t F_PL = (size_t)BB * HD;           // 655,360
    const size_t F_H  = (size_t)BB * BOT;          // 8192
    const size_t fused_need = (F_SC + F_CM + F_CS + F_PT + F_PL + F_H) * sizeof(float);

    if (ws_size >= fused_need) {
        // ---- single pass over x ----
        float* scores = ws;
        float* cmax   = scores + F_SC;
        float* csum   = cmax + F_CM;
        float* part   = csum + F_CS;
        float* pooled = part + F_PT;
        float* h      = pooled + F_PL;

        fused_attn_kernel<<<dim3(NC, BB), 256, 0, stream>>>(x, aw, scores, part, cmax, csum);
        finalize_kernel<<<BB, 256, 0, stream>>>(scores, part, cmax, csum, w_out, pooled);
        mlp1_wmma_kernel<<<dim3(BB / 16, BOT / 16), 32, 0, stream>>>(pooled, w1, b1, h);
        mlp2_kernel<<<BB, 32, 0, stream>>>(h, w2, b2, out_p);
    } else {
        // ---- two passes over x (3 MB workspace) ----
        float* scores = ws;                                  // 65536
        float* w_ws   = ws + (size_t)BB * SS;                // 65536
        float* pooled = w_ws + (size_t)BB * SS;              // 655360
        float* h      = pooled + (size_t)BB * HD;            // 8192

        scores_kernel<<<(BB * SS) / 8, 256, 0, stream>>>(x, aw, scores);
        softmax_kernel<<<BB, SS, 0, stream>>>(scores, w_ws, w_out);
        pooled_kernel<<<dim3(HD / 1024, BB), 256, 0, stream>>>(x, w_ws, pooled);
        mlp1_wmma_kernel<<<dim3(BB / 16, BOT / 16), 32, 0, stream>>>(pooled, w1, b1, h);
        mlp2_kernel<<<BB, 32, 0, stream>>>(h, w2, b2, out_p);
    }
}